// ODEBlock_20993800143010
// MI455X (gfx1250) — compile-verified
//
#include <hip/hip_runtime.h>
#include <math.h>

// ---------------------------------------------------------------------------
// Neural-ODE block (DOPRI5, 48 fixed iterations) on gfx1250.
// f(y) = tanh(y@W1 + b1) @ W2 + b2,  y: 1024x512, W1: 512x1024, W2: 1024x512.
// GEMMs on V_WMMA_F32_16X16X4_F32 (exact fp32 matrix path). Tiles are staged
// by GLOBAL_LOAD_ASYNC_TO_LDS_B128 (ASYNCcnt) in a double-buffered pipeline:
// DMA for chunk c+1 overlaps WMMA compute on chunk c; one barrier per chunk.
// Adaptive-step control state lives in device scalars -> graph-capturable.
// ---------------------------------------------------------------------------

typedef __attribute__((ext_vector_type(2))) float v2f;
typedef __attribute__((ext_vector_type(8))) float v8f;

#define BATCH     1024
#define DATA_DIM  512
#define HIDDEN    1024
#define N_ELEM    (BATCH * DATA_DIM)     // 524288 floats per state vector
#define N4        (N_ELEM / 4)           // 131072 float4
#define MAX_STEPS 48
#define TOLV      1e-3f
#define DT0       0.05f
#define RED_BLOCKS 512

// Recover the 32-bit LDS (addrspace-3) byte offset from a generic pointer:
// flat LDS addresses are {SHARED_BASE, offset[31:0]}, so truncation works.
__device__ __forceinline__ unsigned lds_off(const void* p) {
    return (unsigned)(uintptr_t)p;
}

// Async DMA: 16 bytes/lane, global -> LDS, tracked by ASYNCcnt.
__device__ __forceinline__ void async_g2l_b128(unsigned lds_byte, const void* gptr) {
    asm volatile("global_load_async_to_lds_b128 %0, %1, off"
                 :: "v"(lds_byte), "v"((unsigned long long)(uintptr_t)gptr)
                 : "memory");
}
__device__ __forceinline__ void wait_asynccnt0() {
    asm volatile("s_wait_asynccnt 0x0" ::: "memory");
}

// ---------------------------------------------------------------------------
// fp32 WMMA GEMM:  C = act(A @ B + bias),  A: MxK row-major, B: KxN row-major.
// Block: 256 threads (8 wave32), 128x128 C tile. Wave grid 4(M) x 2(N);
// each wave computes 32x64 = 2x4 WMMA 16x16 tiles.
// K staged through double-buffered LDS in chunks of 16 (4 WMMA k-steps of 4).
//
// ISA fragment layouts (32-bit, wave32):
//   A 16x4 : lane m(0-15) half h : VGPR v = A[m][2h+v]
//   B 4x16 : lane n(0-15) half h : VGPR v = B[2h+v][n]
//   C 16x16: lane n(0-15) half h : VGPR v = C[v+8h][n]
// ---------------------------------------------------------------------------
template<bool TANH>
__global__ __launch_bounds__(256)
void wmma_gemm(const float* __restrict__ A, const float* __restrict__ B,
               const float* __restrict__ bias, float* __restrict__ C,
               int M, int N, int K)
{
    constexpr int AP  = 20;        // A tile pitch: 80B rows (B128-aligned)
    constexpr int BP  = 136;       // B tile pitch: 544B rows (B128-aligned)
    constexpr int ASZ = 128 * AP;  // floats per A buffer
    constexpr int BSZ = 16 * BP;   // floats per B buffer
    __shared__ float As[2 * ASZ];
    __shared__ float Bs[2 * BSZ];

    const int tid   = threadIdx.x;
    const int lane  = tid & 31;
    const int wid   = tid >> 5;
    const int half  = (lane >> 4) & 1;   // K-half selector within fragments
    const int mn    = lane & 15;         // M (for A) / N (for B,C) lane index
    const int waveM = (wid & 3) * 32;
    const int waveN = (wid >> 2) * 64;
    const int m0    = blockIdx.y * 128;
    const int n0    = blockIdx.x * 128;

    // Per-thread staging coordinates (2 x B128 per tile per thread).
    const int i0  = tid * 2;
    const int ar0 = (i0 + 0) >> 2, ac0 = ((i0 + 0) & 3) * 4;
    const int ar1 = (i0 + 1) >> 2, ac1 = ((i0 + 1) & 3) * 4;
    const int br0 = (i0 + 0) >> 5, bc0 = ((i0 + 0) & 31) * 4;
    const int br1 = (i0 + 1) >> 5, bc1 = ((i0 + 1) & 31) * 4;
    const unsigned aOff0 = lds_off(&As[ar0 * AP + ac0]);
    const unsigned aOff1 = lds_off(&As[ar1 * AP + ac1]);
    const unsigned bOff0 = lds_off(&Bs[br0 * BP + bc0]);
    const unsigned bOff1 = lds_off(&Bs[br1 * BP + bc1]);

    // Issue the 4 async B128 copies for K-chunk k0 into LDS buffer b,
    // and prefetch the chunk after it toward L2.
    auto issue = [&](int k0, int b) {
        const unsigned ab = (unsigned)b * (ASZ * 4u);
        const unsigned bb = (unsigned)b * (BSZ * 4u);
        async_g2l_b128(aOff0 + ab, &A[(size_t)(m0 + ar0) * K + k0 + ac0]);
        async_g2l_b128(aOff1 + ab, &A[(size_t)(m0 + ar1) * K + k0 + ac1]);
        async_g2l_b128(bOff0 + bb, &B[(size_t)(k0 + br0) * N + n0 + bc0]);
        async_g2l_b128(bOff1 + bb, &B[(size_t)(k0 + br1) * N + n0 + bc1]);
        if (k0 + 16 < K) {
            __builtin_prefetch(&A[(size_t)(m0 + ar0) * K + k0 + 16 + ac0], 0, 0);
            __builtin_prefetch(&B[(size_t)(k0 + 16 + br0) * N + n0 + bc0], 0, 0);
        }
    };

    v8f acc[2][4];
#pragma unroll
    for (int mt = 0; mt < 2; ++mt)
#pragma unroll
        for (int nt = 0; nt < 4; ++nt)
            acc[mt][nt] = (v8f){0.f,0.f,0.f,0.f,0.f,0.f,0.f,0.f};

    // ---- pipeline prologue: DMA chunk 0 into buffer 0
    issue(0, 0);

    const int nchunk = K >> 4;
    for (int c = 0; c < nchunk; ++c) {
        const int cur = c & 1;
        wait_asynccnt0();       // this wave's DMA for chunk c complete
        __syncthreads();        // all waves' chunk-c data published;
                                // also: everyone is done reading buffer cur^1
        if (c + 1 < nchunk) issue((c + 1) << 4, cur ^ 1);

        const float* __restrict__ Ab = &As[cur * ASZ];
        const float* __restrict__ Bb = &Bs[cur * BSZ];

#pragma unroll
        for (int ks = 0; ks < 4; ++ks) {
            const int kk = ks * 4 + 2 * half;   // this lane-half's K base
            v2f af[2], bf[4];
#pragma unroll
            for (int mt = 0; mt < 2; ++mt) {
                const int row = waveM + mt * 16 + mn;
                af[mt][0] = Ab[row * AP + kk + 0];
                af[mt][1] = Ab[row * AP + kk + 1];
            }
#pragma unroll
            for (int nt = 0; nt < 4; ++nt) {
                const int col = waveN + nt * 16 + mn;
                bf[nt][0] = Bb[(kk + 0) * BP + col];
                bf[nt][1] = Bb[(kk + 1) * BP + col];
            }
#pragma unroll
            for (int mt = 0; mt < 2; ++mt)
#pragma unroll
                for (int nt = 0; nt < 4; ++nt)
                    acc[mt][nt] = __builtin_amdgcn_wmma_f32_16x16x4_f32(
                        false, af[mt], false, bf[nt],
                        (short)0, acc[mt][nt], false, false);
        }
    }

    // ---- epilogue: bias (+ tanh), write C row-major
#pragma unroll
    for (int mt = 0; mt < 2; ++mt) {
#pragma unroll
        for (int nt = 0; nt < 4; ++nt) {
            const int col = n0 + waveN + nt * 16 + mn;
            const float b = bias[col];
#pragma unroll
            for (int v = 0; v < 8; ++v) {
                const int row = m0 + waveM + mt * 16 + half * 8 + v;
                float val = acc[mt][nt][v] + b;
                if constexpr (TANH) val = tanhf(val);
                C[(size_t)row * N + col] = val;
            }
        }
    }
}

// ---------------------------------------------------------------------------
// Elementwise stage combination: out = y + dt_c * sum_j c_j * k_j
// ---------------------------------------------------------------------------
__device__ inline float4 f4_fma(float a, float4 v, float4 acc) {
    acc.x = fmaf(a, v.x, acc.x); acc.y = fmaf(a, v.y, acc.y);
    acc.z = fmaf(a, v.z, acc.z); acc.w = fmaf(a, v.w, acc.w);
    return acc;
}

template<int NK>
__global__ __launch_bounds__(256)
void stage_combine(const float4* __restrict__ y,
                   const float4* __restrict__ ka, const float4* __restrict__ kb,
                   const float4* __restrict__ kc, const float4* __restrict__ kd,
                   const float4* __restrict__ ke,
                   float ca, float cb, float cc, float cd, float ce,
                   const float* __restrict__ scal, float4* __restrict__ out)
{
    const int i = blockIdx.x * blockDim.x + threadIdx.x;
    if (i >= N4) return;
    const float dtc = scal[2];
    float4 s = {0.f, 0.f, 0.f, 0.f};
    s = f4_fma(ca, ka[i], s);
    if constexpr (NK >= 2) s = f4_fma(cb, kb[i], s);
    if constexpr (NK >= 3) s = f4_fma(cc, kc[i], s);
    if constexpr (NK >= 4) s = f4_fma(cd, kd[i], s);
    if constexpr (NK >= 5) s = f4_fma(ce, ke[i], s);
    float4 yy = y[i], o;
    o.x = fmaf(dtc, s.x, yy.x); o.y = fmaf(dtc, s.y, yy.y);
    o.z = fmaf(dtc, s.z, yy.z); o.w = fmaf(dtc, s.w, yy.w);
    out[i] = o;
}

// ---------------------------------------------------------------------------
// Error-norm partial reduction: sum over ((dt_c * e·k) / scale)^2
// err coeffs over k1,k3,k4,k5,k6,k7 (e2 = 0).
// ---------------------------------------------------------------------------
__global__ __launch_bounds__(256)
void err_reduce(const float4* __restrict__ k1, const float4* __restrict__ k3,
                const float4* __restrict__ k4, const float4* __restrict__ k5,
                const float4* __restrict__ k6, const float4* __restrict__ k7,
                const float4* __restrict__ y,  const float4* __restrict__ y5,
                const float* __restrict__ scal, float* __restrict__ partial)
{
    const float E1 =  71.0f / 57600.0f;
    const float E3 = -71.0f / 16695.0f;
    const float E4 =  71.0f / 1920.0f;
    const float E5 = -17253.0f / 339200.0f;
    const float E6 =  22.0f / 525.0f;
    const float E7 = -1.0f / 40.0f;

    __shared__ float sh[256];
    const int i = blockIdx.x * blockDim.x + threadIdx.x;
    const float dtc = scal[2];
    float s = 0.f;
    if (i < N4) {
        float4 a = k1[i], c = k3[i], d = k4[i], e = k5[i], f = k6[i], g = k7[i];
        float4 yy = y[i], y2 = y5[i];
        float ex, sc, r;
        ex = dtc * (E1*a.x + E3*c.x + E4*d.x + E5*e.x + E6*f.x + E7*g.x);
        sc = TOLV + TOLV * fmaxf(fabsf(yy.x), fabsf(y2.x)); r = ex / sc; s += r * r;
        ex = dtc * (E1*a.y + E3*c.y + E4*d.y + E5*e.y + E6*f.y + E7*g.y);
        sc = TOLV + TOLV * fmaxf(fabsf(yy.y), fabsf(y2.y)); r = ex / sc; s += r * r;
        ex = dtc * (E1*a.z + E3*c.z + E4*d.z + E5*e.z + E6*f.z + E7*g.z);
        sc = TOLV + TOLV * fmaxf(fabsf(yy.z), fabsf(y2.z)); r = ex / sc; s += r * r;
        ex = dtc * (E1*a.w + E3*c.w + E4*d.w + E5*e.w + E6*f.w + E7*g.w);
        sc = TOLV + TOLV * fmaxf(fabsf(yy.w), fabsf(y2.w)); r = ex / sc; s += r * r;
    }
    sh[threadIdx.x] = s;
    __syncthreads();
    for (int off = 128; off > 0; off >>= 1) {
        if (threadIdx.x < off) sh[threadIdx.x] += sh[threadIdx.x + off];
        __syncthreads();
    }
    if (threadIdx.x == 0) partial[blockIdx.x] = sh[0];
}

// ---------------------------------------------------------------------------
// Step-begin: dt_c = min(dt, 1 - t), done = (t >= 1)
// scal layout: [0]=t [1]=dt [2]=dt_c [3]=done [4]=upd
// ---------------------------------------------------------------------------
__global__ void step_begin(float* scal)
{
    if (threadIdx.x == 0 && blockIdx.x == 0) {
        const float t = scal[0], dt = scal[1];
        scal[3] = (t >= 1.0f) ? 1.0f : 0.0f;
        scal[2] = fminf(dt, 1.0f - t);
    }
}

__global__ void init_scalars(float* scal)
{
    if (threadIdx.x == 0 && blockIdx.x == 0) {
        scal[0] = 0.0f;   // t
        scal[1] = DT0;    // dt
        scal[2] = DT0;    // dt_c
        scal[3] = 0.0f;   // done
        scal[4] = 0.0f;   // upd
    }
}

// ---------------------------------------------------------------------------
// Final reduction + step control: err_norm, accept, fac, t/dt update.
// ---------------------------------------------------------------------------
__global__ __launch_bounds__(256)
void control_kernel(const float* __restrict__ partial, float* scal)
{
    __shared__ float sh[256];
    float s = 0.f;
    for (int i = threadIdx.x; i < RED_BLOCKS; i += 256) s += partial[i];
    sh[threadIdx.x] = s;
    __syncthreads();
    for (int off = 128; off > 0; off >>= 1) {
        if (threadIdx.x < off) sh[threadIdx.x] += sh[threadIdx.x + off];
        __syncthreads();
    }
    if (threadIdx.x == 0) {
        const float total = sh[0];
        float err_norm = fmaxf(sqrtf(total * (1.0f / (float)N_ELEM)), 1e-10f);
        const float t = scal[0], dt = scal[1], dtc = scal[2];
        const bool  isdone = scal[3] != 0.0f;
        const bool  accept = err_norm <= 1.0f;
        float fac = 0.9f * powf(err_norm, -0.2f);
        fac = fminf(fmaxf(fac, 0.2f), 10.0f);
        const bool upd = accept && !isdone;
        scal[0] = upd ? (t + dtc) : t;
        scal[1] = isdone ? dt : (dtc * fac);
        scal[4] = upd ? 1.0f : 0.0f;
    }
}

// ---------------------------------------------------------------------------
// Accept update (FSAL): if upd { y = y5; k1 = k7; }
// ---------------------------------------------------------------------------
__global__ __launch_bounds__(256)
void accept_update(float4* __restrict__ y, float4* __restrict__ k1,
                   const float4* __restrict__ y5, const float4* __restrict__ k7,
                   const float* __restrict__ scal)
{
    const int i = blockIdx.x * blockDim.x + threadIdx.x;
    if (i >= N4) return;
    if (scal[4] != 0.0f) {
        y[i]  = y5[i];
        k1[i] = k7[i];
    }
}

// ---------------------------------------------------------------------------
extern "C" void kernel_launch(void* const* d_in, const int* in_sizes, int n_in,
                              void* d_out, int out_size, void* d_ws, size_t ws_size,
                              hipStream_t stream)
{
    (void)in_sizes; (void)n_in; (void)out_size; (void)ws_size;

    const float* x  = (const float*)d_in[0];   // 1024x512
    const float* W1 = (const float*)d_in[1];   // 512x1024
    const float* b1 = (const float*)d_in[2];   // 1024
    const float* W2 = (const float*)d_in[3];   // 1024x512
    const float* b2 = (const float*)d_in[4];   // 512

    float* ws = (float*)d_ws;
    float* scal    = ws;            // 16 floats of control scalars
    float* partial = ws + 16;       // RED_BLOCKS partials
    size_t off = 1024;
    float* y   = ws + off; off += N_ELEM;
    float* y5  = ws + off; off += N_ELEM;
    float* yst = ws + off; off += N_ELEM;
    float* k[7];
    for (int j = 0; j < 7; ++j) { k[j] = ws + off; off += N_ELEM; }
    float* H   = ws + off; off += (size_t)BATCH * HIDDEN;

    const dim3 blk(256);
    const dim3 g1(HIDDEN / 128, BATCH / 128);     // GEMM1: C 1024x1024
    const dim3 g2(DATA_DIM / 128, BATCH / 128);   // GEMM2: C 1024x512
    const dim3 gv(RED_BLOCKS);                    // 512*256 == N4 exactly

    auto f_eval = [&](const float* in, float* out_k) {
        wmma_gemm<true ><<<g1, blk, 0, stream>>>(in, W1, b1, H, BATCH, HIDDEN, DATA_DIM);
        wmma_gemm<false><<<g2, blk, 0, stream>>>(H,  W2, b2, out_k, BATCH, DATA_DIM, HIDDEN);
    };

    // ---- init: t=0, dt=DT0, y = x, k1 = f(x)
    init_scalars<<<1, 1, 0, stream>>>(scal);
    hipMemcpyAsync(y, x, (size_t)N_ELEM * sizeof(float),
                   hipMemcpyDeviceToDevice, stream);
    f_eval(x, k[0]);

    const float4* y4  = (const float4*)y;
    const float4* k14 = (const float4*)k[0];
    const float4* k24 = (const float4*)k[1];
    const float4* k34 = (const float4*)k[2];
    const float4* k44 = (const float4*)k[3];
    const float4* k54 = (const float4*)k[4];
    const float4* k64 = (const float4*)k[5];
    const float4* k74 = (const float4*)k[6];
    float4* yst4 = (float4*)yst;
    float4* y54  = (float4*)y5;

    for (int it = 0; it < MAX_STEPS; ++it) {
        step_begin<<<1, 1, 0, stream>>>(scal);

        // k2 = f(y + dt*(0.2 k1))
        stage_combine<1><<<gv, blk, 0, stream>>>(y4, k14, k14, k14, k14, k14,
            0.2f, 0.f, 0.f, 0.f, 0.f, scal, yst4);
        f_eval(yst, k[1]);

        // k3 = f(y + dt*(3/40 k1 + 9/40 k2))
        stage_combine<2><<<gv, blk, 0, stream>>>(y4, k14, k24, k14, k14, k14,
            3.0f/40.0f, 9.0f/40.0f, 0.f, 0.f, 0.f, scal, yst4);
        f_eval(yst, k[2]);

        // k4 = f(y + dt*(44/45 k1 - 56/15 k2 + 32/9 k3))
        stage_combine<3><<<gv, blk, 0, stream>>>(y4, k14, k24, k34, k14, k14,
            44.0f/45.0f, -56.0f/15.0f, 32.0f/9.0f, 0.f, 0.f, scal, yst4);
        f_eval(yst, k[3]);

        // k5
        stage_combine<4><<<gv, blk, 0, stream>>>(y4, k14, k24, k34, k44, k14,
            19372.0f/6561.0f, -25360.0f/2187.0f, 64448.0f/6561.0f,
            -212.0f/729.0f, 0.f, scal, yst4);
        f_eval(yst, k[4]);

        // k6
        stage_combine<5><<<gv, blk, 0, stream>>>(y4, k14, k24, k34, k44, k54,
            9017.0f/3168.0f, -355.0f/33.0f, 46732.0f/5247.0f,
            49.0f/176.0f, -5103.0f/18656.0f, scal, yst4);
        f_eval(yst, k[5]);

        // y5 = y + dt*(35/384 k1 + 500/1113 k3 + 125/192 k4 - 2187/6784 k5 + 11/84 k6)
        stage_combine<5><<<gv, blk, 0, stream>>>(y4, k14, k34, k44, k54, k64,
            35.0f/384.0f, 500.0f/1113.0f, 125.0f/192.0f,
            -2187.0f/6784.0f, 11.0f/84.0f, scal, y54);
        f_eval(y5, k[6]);   // k7 = f(y5), FSAL

        err_reduce<<<gv, blk, 0, stream>>>(k14, k34, k44, k54, k64, k74,
                                           y4, (const float4*)y5, scal, partial);
        control_kernel<<<1, blk, 0, stream>>>(partial, scal);
        accept_update<<<gv, blk, 0, stream>>>((float4*)y, (float4*)k[0],
                                              (const float4*)y5, k74, scal);
    }

    hipMemcpyAsync(d_out, y, (size_t)N_ELEM * sizeof(float),
                   hipMemcpyDeviceToDevice, stream);
}